// UserGNNEncoder_74603581931928
// MI455X (gfx1250) — compile-verified
//
#include <hip/hip_runtime.h>

// ---------------------------------------------------------------------------
// GATv2 hetero-GNN encoder for MI455X (gfx1250, wave32).
//
// Roofline: dense transforms are ~3 GFLOP of 64x64 GEMMs (negligible; done
// with v_wmma_f32_16x16x4_f32, fp32 like the reference). The edge phase moves
// ~4 GB of 256B-granule gather/scatter traffic across ~4.05M edges; node
// tables (12.8-25.6 MB) are L2-resident (192 MB L2), so this is L2-bw /
// atomic bound. Wave-per-edge with one float2 (b64) load per lane gives one
// fully-coalesced 256B transaction per row access.
//
// GEMM: persistent blocks, W (16KB) staged in LDS once per block, grid-stride
// over 16-row stripes; 4 waves/block each own one 16x16 N-tile via 16 chained
// K=4 WMMAs. LDS rows padded to 65 floats -> conflict-free column reads.
// ---------------------------------------------------------------------------

typedef float v8f __attribute__((ext_vector_type(8)));
typedef float v2f __attribute__((ext_vector_type(2)));

#define NEG_SLOPE 0.2f

// ---------------------------------------------------------------------------
// Utility kernels
// ---------------------------------------------------------------------------
__global__ __launch_bounds__(256) void fill_f32(float* p, float v, long long n) {
  long long i = (long long)blockIdx.x * blockDim.x + threadIdx.x;
  long long stride = (long long)gridDim.x * blockDim.x;
  for (; i < n; i += stride) p[i] = v;
}

// x[i] = relu(x[i] + bias[i % 64]), in place (node features are 64-wide)
__global__ __launch_bounds__(256) void bias_relu_inplace(float* x, const float* __restrict__ bias,
                                                         long long n) {
  long long i = (long long)blockIdx.x * blockDim.x + threadIdx.x;
  if (i < n) {
    float v = x[i] + bias[i & 63];
    x[i] = v > 0.f ? v : 0.f;
  }
}

// ---------------------------------------------------------------------------
// Y[M,64] = X[M,64] @ W[64,64] (+ bias), via V_WMMA_F32_16X16X4_F32.
//
// WMMA f32 16x16x4 operand layout (wave32):
//   A (16x4):  lane L holds row M=L&15; VGPR0/1 = K {0,1} (lanes 0-15) or
//              K {2,3} (lanes 16-31).
//   B (4x16):  lane L holds col N=L&15; VGPR0/1 = K {0,1} / {2,3} likewise.
//   C/D:       VGPR r -> M = r + 8*(lane>=16), N = lane&15.
// ---------------------------------------------------------------------------
__global__ __launch_bounds__(128) void gemm64_wmma(const float* __restrict__ X,
                                                   const float* __restrict__ W,
                                                   const float* __restrict__ bias,
                                                   float* __restrict__ Y, int M, int relu) {
  __shared__ float Ws[64][65];
  __shared__ float Xs[16][65];
  __shared__ float Bs[64];

  const int tid = threadIdx.x;
  const int ntile = (M + 15) >> 4;

  // Stage W once per (persistent) block, b128 global loads.
  for (int i = tid; i < 1024; i += 128) {
    const float4 v = ((const float4*)W)[i];
    const int r = i >> 4, c = (i & 15) * 4;
    Ws[r][c] = v.x;
    Ws[r][c + 1] = v.y;
    Ws[r][c + 2] = v.z;
    Ws[r][c + 3] = v.w;
  }
  if (tid < 64) Bs[tid] = bias ? bias[tid] : 0.f;

  const int wave = tid >> 5;
  const int lane = tid & 31;
  const int half = lane >> 4;      // 0: lanes 0-15, 1: lanes 16-31
  const int l16 = lane & 15;
  const int kbase = half * 2;      // K offset within each K=4 chunk
  const int ncol = wave * 16 + l16;

  for (int t = blockIdx.x; t < ntile; t += gridDim.x) {
    const int row0 = t * 16;

    __syncthreads();  // Ws ready (1st iter) / previous iter's readers done
    for (int i = tid; i < 256; i += 128) {
      const int r = i >> 4, c = (i & 15) * 4;
      const int gr = row0 + r;
      float4 v = {0.f, 0.f, 0.f, 0.f};
      if (gr < M) v = ((const float4*)X)[gr * 16 + (i & 15)];
      Xs[r][c] = v.x;
      Xs[r][c + 1] = v.y;
      Xs[r][c + 2] = v.z;
      Xs[r][c + 3] = v.w;
    }
    __syncthreads();

    v8f c = {};
#pragma unroll
    for (int k0 = 0; k0 < 16; ++k0) {
      const int k = k0 * 4 + kbase;
      v2f a, b;
      a.x = Xs[l16][k];
      a.y = Xs[l16][k + 1];
      b.x = Ws[k][ncol];
      b.y = Ws[k + 1][ncol];
      c = __builtin_amdgcn_wmma_f32_16x16x4_f32(false, a, false, b, (short)0, c, false, false);
    }

#pragma unroll
    for (int r = 0; r < 8; ++r) {
      const int m = r + half * 8;
      const int gr = row0 + m;
      if (gr < M) {
        float v = c[r] + Bs[ncol];
        if (relu) v = v > 0.f ? v : 0.f;
        Y[(long long)gr * 64 + ncol] = v;
      }
    }
  }
}

// ---------------------------------------------------------------------------
// Edge kernels (wave-per-edge; each lane owns features {2*lane, 2*lane+1},
// loaded as one float2 -> a single coalesced 256B row transaction per wave).
// Edges with id >= n_real are synthesized self-loops (src=dst=id-n_real).
// ---------------------------------------------------------------------------
__device__ inline void atomic_max_f32(float* addr, float val) {
  int* ia = (int*)addr;
  int cur = __float_as_int(*addr);
  while (__int_as_float(cur) < val) {
    int prev = atomicCAS(ia, cur, __float_as_int(val));
    if (prev == cur) break;
    cur = prev;
  }
}

// Pass 1: e[edge] = att . leakyrelu(hl[src] + hr[dst] (+ label*We)), and
//         segment max via atomic max on mmax[dst].
__global__ __launch_bounds__(256) void edge_e_kernel(
    const float* __restrict__ hl, const float* __restrict__ hr, const int* __restrict__ src,
    const int* __restrict__ dst, const float* __restrict__ att, const float* __restrict__ elabel,
    const float* __restrict__ We, float* __restrict__ e_out, float* __restrict__ mmax, int n_real,
    int n_total) {
  const int wid = (int)(((long long)blockIdx.x * blockDim.x + threadIdx.x) >> 5);
  const int lane = threadIdx.x & 31;
  if (wid >= n_total) return;

  int s, d;
  float lab = 0.f;
  if (wid < n_real) {
    s = src[wid];
    d = dst[wid];
    if (elabel) lab = elabel[wid];
  } else {
    s = wid - n_real;
    d = s;
  }

  const float2 a2 = ((const float2*)(hl + (long long)s * 64))[lane];
  const float2 b2 = ((const float2*)(hr + (long long)d * 64))[lane];
  const float2 t2 = ((const float2*)att)[lane];

  float m0 = a2.x + b2.x;
  float m1 = a2.y + b2.y;
  if (We) {
    const float2 w2 = ((const float2*)We)[lane];
    m0 += lab * w2.x;
    m1 += lab * w2.y;
  }
  m0 = m0 > 0.f ? m0 : NEG_SLOPE * m0;
  m1 = m1 > 0.f ? m1 : NEG_SLOPE * m1;
  float acc = m0 * t2.x + m1 * t2.y;

#pragma unroll
  for (int off = 16; off; off >>= 1) acc += __shfl_xor(acc, off, 32);

  if (lane == 0) {
    e_out[wid] = acc;
    atomic_max_f32(&mmax[d], acc);
  }
}

// Pass 2: w[edge] = exp(e - mmax[dst]); ssum[dst] += w (atomic).
__global__ __launch_bounds__(256) void edge_w_kernel(const float* __restrict__ e_in,
                                                     const int* __restrict__ dst,
                                                     const float* __restrict__ mmax,
                                                     float* __restrict__ ssum,
                                                     float* __restrict__ w_out, int n_real,
                                                     int n_total) {
  const int i = (int)((long long)blockIdx.x * blockDim.x + threadIdx.x);
  if (i >= n_total) return;
  const int d = (i < n_real) ? dst[i] : (i - n_real);
  const float w = __expf(e_in[i] - mmax[d]);
  w_out[i] = w;
  atomicAdd(&ssum[d], w);
}

// Pass 3: acc[dst] += (w / (ssum[dst]+eps)) * hl[src]  (64-wide atomic scatter)
__global__ __launch_bounds__(256) void edge_scatter_kernel(
    const float* __restrict__ hl, const float* __restrict__ w_in, const float* __restrict__ ssum,
    const int* __restrict__ src, const int* __restrict__ dst, float* __restrict__ acc, int n_real,
    int n_total) {
  const int wid = (int)(((long long)blockIdx.x * blockDim.x + threadIdx.x) >> 5);
  const int lane = threadIdx.x & 31;
  if (wid >= n_total) return;

  int s, d;
  if (wid < n_real) {
    s = src[wid];
    d = dst[wid];
  } else {
    s = wid - n_real;
    d = s;
  }

  const float coef = w_in[wid] / (ssum[d] + 1e-16f);
  const float2 v2 = ((const float2*)(hl + (long long)s * 64))[lane];
  float* out = acc + (long long)d * 64 + lane * 2;
  atomicAdd(&out[0], coef * v2.x);
  atomicAdd(&out[1], coef * v2.y);
}

// ---------------------------------------------------------------------------
// Host-side orchestration
// ---------------------------------------------------------------------------
extern "C" void kernel_launch(void* const* d_in, const int* in_sizes, int n_in, void* d_out,
                              int out_size, void* d_ws, size_t ws_size, hipStream_t stream) {
  // ---- input pointers (setup_inputs dict order) ----
  const float* x_product = (const float*)d_in[0];
  const float* x_customer = (const float*)d_in[1];
  const float* edge_label = (const float*)d_in[2];
  const int* src_pp = (const int*)d_in[3];
  const int* dst_pp = (const int*)d_in[4];
  const int* src_pc = (const int*)d_in[5];
  const int* dst_pc = (const int*)d_in[6];
  const float* W1l = (const float*)d_in[7];
  const float* b1l = (const float*)d_in[8];
  const float* W1r = (const float*)d_in[9];
  const float* b1r = (const float*)d_in[10];
  const float* att1 = (const float*)d_in[11];
  const float* bias1 = (const float*)d_in[12];
  const float* W2l = (const float*)d_in[13];
  const float* b2l = (const float*)d_in[14];
  const float* W2r = (const float*)d_in[15];
  const float* b2r = (const float*)d_in[16];
  const float* We2 = (const float*)d_in[17];
  const float* att2 = (const float*)d_in[18];
  const float* bias2 = (const float*)d_in[19];
  const float* W3l = (const float*)d_in[20];
  const float* b3l = (const float*)d_in[21];
  const float* W3r = (const float*)d_in[22];
  const float* b3r = (const float*)d_in[23];
  const float* We3 = (const float*)d_in[24];
  const float* att3 = (const float*)d_in[25];
  const float* bias3 = (const float*)d_in[26];
  const float* Wlin = (const float*)d_in[27];
  const float* blin = (const float*)d_in[28];

  const long long P = in_sizes[0] / 64;   // products
  const long long C = in_sizes[1] / 64;   // customers
  const long long EPP = in_sizes[3];
  const long long EPC = in_sizes[5];
  const long long E1 = EPP + P;           // conv1 edges incl. self loops
  const long long NMAX = (P > C) ? P : C;
  const long long EMAX = (E1 > EPC) ? E1 : EPC;

  // ---- workspace carve-up (float units, 256B aligned) ----
  float* ws = (float*)d_ws;
  size_t off = 0;
  auto carve = [&](long long nfloats) -> float* {
    float* p = ws + off;
    off += (size_t)((nfloats + 63) & ~63LL);
    return p;
  };
  float* HL = carve(P * 64);       // reused: hl1, hl2, hl3
  float* HR = carve(NMAX * 64);    // reused: hr1, hr2, hr3
  float* EB = carve(EMAX);         // per-edge logit e
  float* WB = carve(EMAX);         // per-edge exp weight
  float* MM = carve(NMAX);         // segment max
  float* SS = carve(NMAX);         // segment sum
  float* ACC1 = carve(P * 64);     // product_x (persists to conv3)
  float* ACC2 = carve(C * 64);     // customer_x after conv2 (persists)
  float* ACC3 = carve(C * 64);     // customer_x after conv3

  const float NEG_INF = -__builtin_huge_valf();
  auto fill = [&](float* p, float v, long long n) {
    int blocks = (int)((n + 255) / 256);
    if (blocks > 16384) blocks = 16384;
    fill_f32<<<blocks, 256, 0, stream>>>(p, v, n);
  };
  auto gemm = [&](const float* X, const float* W, const float* b, float* Y, long long M,
                  int relu) {
    int ntile = (int)((M + 15) / 16);
    int blocks = ntile < 2048 ? ntile : 2048;  // persistent blocks, W staged once each
    gemm64_wmma<<<blocks, 128, 0, stream>>>(X, W, b, Y, (int)M, relu);
  };
  auto run_edges = [&](const float* hl, const float* hr, const int* src, const int* dst,
                       const float* att, const float* elab, const float* We, float* acc,
                       long long n_real, long long n_total, long long n_dst) {
    fill(MM, NEG_INF, n_dst);
    fill(SS, 0.f, n_dst);
    fill(acc, 0.f, n_dst * 64);
    const int wpb = 8;  // 8 waves (edges) per 256-thread block
    edge_e_kernel<<<(int)((n_total + wpb - 1) / wpb), 256, 0, stream>>>(
        hl, hr, src, dst, att, elab, We, EB, MM, (int)n_real, (int)n_total);
    edge_w_kernel<<<(int)((n_total + 255) / 256), 256, 0, stream>>>(EB, dst, MM, SS, WB,
                                                                    (int)n_real, (int)n_total);
    edge_scatter_kernel<<<(int)((n_total + wpb - 1) / wpb), 256, 0, stream>>>(
        hl, WB, SS, src, dst, acc, (int)n_real, (int)n_total);
  };

  // ---- conv1: product -> product (self loops, no edge features) ----
  gemm(x_product, W1l, b1l, HL, P, 0);
  gemm(x_product, W1r, b1r, HR, P, 0);
  run_edges(HL, HR, src_pp, dst_pp, att1, nullptr, nullptr, ACC1, EPP, E1, P);
  bias_relu_inplace<<<(int)((P * 64 + 255) / 256), 256, 0, stream>>>(ACC1, bias1, P * 64);

  // ---- conv2: product -> customer (edge features = label * We2) ----
  gemm(x_product, W2l, b2l, HL, P, 0);
  gemm(x_customer, W2r, b2r, HR, C, 0);
  run_edges(HL, HR, src_pc, dst_pc, att2, edge_label, We2, ACC2, EPC, EPC, C);
  bias_relu_inplace<<<(int)((C * 64 + 255) / 256), 256, 0, stream>>>(ACC2, bias2, C * 64);

  // ---- conv3: hidden product -> hidden customer ----
  gemm(ACC1, W3l, b3l, HL, P, 0);
  gemm(ACC2, W3r, b3r, HR, C, 0);
  run_edges(HL, HR, src_pc, dst_pc, att3, edge_label, We3, ACC3, EPC, EPC, C);
  bias_relu_inplace<<<(int)((C * 64 + 255) / 256), 256, 0, stream>>>(ACC3, bias3, C * 64);

  // ---- final linear: out = customer_x @ Wlin + blin ----
  gemm(ACC3, Wlin, blin, (float*)d_out, C, 0);
}